// LocalPixelShuffling_65206193488193
// MI455X (gfx1250) — compile-verified
//
#include <hip/hip_runtime.h>
#include <stdint.h>

// ---------------------------------------------------------------------------
// LocalPixelShuffling, MI455X (gfx1250).
//  - Bulk copy orig->out via Tensor Data Mover (tensor_load_to_lds /
//    tensor_store_from_lds + s_wait_tensorcnt), ~402 MB => ~17us @ 23.3 TB/s.
//  - Window shuffles fully parallel (one block per window) with
//    last-writer-wins conflict resolution instead of 1000 serial steps.
//  - JAX threefry2x32 PRNG replicated bit-exactly (split / randint / uniform).
// ---------------------------------------------------------------------------

#define Cc 3
#define Hh 4096
#define Ww 4096
#define NW 1000
#define MB 8            // MAX_BLOCK
#define TROWS 4         // rows per TDM tile: 4 * 4096 * 4B = 64 KB LDS

typedef unsigned int u32;
typedef u32 v4u __attribute__((ext_vector_type(4)));
typedef int  v8i __attribute__((ext_vector_type(8)));
typedef int  v4i __attribute__((ext_vector_type(4)));

// ---- JAX threefry2x32 (20 rounds, standard rotation schedule) -------------
__device__ __forceinline__ void threefry2x32(u32 k0, u32 k1, u32 x0, u32 x1,
                                             u32& o0, u32& o1) {
  u32 ks2 = k0 ^ k1 ^ 0x1BD11BDAu;
  x0 += k0; x1 += k1;
#define TF_R(r) { x0 += x1; x1 = (x1 << (r)) | (x1 >> (32 - (r))); x1 ^= x0; }
  TF_R(13) TF_R(15) TF_R(26) TF_R(6)
  x0 += k1;  x1 += ks2 + 1u;
  TF_R(17) TF_R(29) TF_R(16) TF_R(24)
  x0 += ks2; x1 += k0 + 2u;
  TF_R(13) TF_R(15) TF_R(26) TF_R(6)
  x0 += k0;  x1 += k1 + 3u;
  TF_R(17) TF_R(29) TF_R(16) TF_R(24)
  x0 += k1;  x1 += ks2 + 4u;
  TF_R(13) TF_R(15) TF_R(26) TF_R(6)
  x0 += ks2; x1 += k0 + 5u;
#undef TF_R
  o0 = x0; o1 = x1;
}

// JAX uniform [0,1): bitcast((bits>>9)|0x3f800000) - 1.0
__device__ __forceinline__ float u01(u32 bits) {
  return __uint_as_float((bits >> 9) | 0x3F800000u) - 1.0f;
}

// ---------------------------------------------------------------------------
// Kernel 0: bandwidth-bound copy of the whole image through the TDM.
// One wave per block; the TDM descriptor describes a TROWS x 4096 fp32 tile.
// ---------------------------------------------------------------------------
__global__ void bulk_copy_tdm(const float* __restrict__ src,
                              float* __restrict__ dst) {
#if __has_builtin(__builtin_amdgcn_tensor_load_to_lds) && \
    __has_builtin(__builtin_amdgcn_tensor_store_from_lds) && \
    __has_builtin(__builtin_amdgcn_s_wait_tensorcnt)
  __shared__ u32 stage[TROWS * Ww];                 // 64 KB staging tile
  u32 lds_off = (u32)(uintptr_t)&stage[0];
  uint64_t byte_off = (uint64_t)blockIdx.x * (uint64_t)(TROWS * Ww * 4);
  uint64_t ga_src = (uint64_t)(uintptr_t)src + byte_off;
  uint64_t ga_dst = (uint64_t)(uintptr_t)dst + byte_off;

  // D# group 0: count=1 (valid), lds_addr, global_addr[56:0], type=2
  v4u g0l = { 1u, lds_off, (u32)ga_src,
              ((u32)((ga_src >> 32) & 0x1FFFFFFu)) | 0x80000000u };
  v4u g0s = { 1u, lds_off, (u32)ga_dst,
              ((u32)((ga_dst >> 32) & 0x1FFFFFFu)) | 0x80000000u };
  // D# group 1: data_size=4B; tensor_dim0=4096, tensor_dim1=TROWS;
  //             tile_dim0=4096, tile_dim1=TROWS; dim0_stride=4096
  v8i g1 = { (int)0x20000u,            // data_size=2 (4 bytes)
             (int)((u32)Ww   << 16),   // tensor_dim0 lo16
             (int)((u32)TROWS << 16),  // tensor_dim0 hi16 | tensor_dim1 lo16
             (int)((u32)Ww   << 16),   // tensor_dim1 hi16 | tile_dim0
             (int)TROWS,               // tile_dim1 | tile_dim2<<16
             (int)Ww,                  // tensor_dim0_stride lo32
             0, 0 };
  v4i gz4 = { 0, 0, 0, 0 };
  v8i gz8 = { 0, 0, 0, 0, 0, 0, 0, 0 };

  __builtin_amdgcn_tensor_load_to_lds(g0l, g1, gz4, gz4, gz8, 0);
  __builtin_amdgcn_s_wait_tensorcnt(0);
  __builtin_amdgcn_tensor_store_from_lds(g0s, g1, gz4, gz4, gz8, 0);
  __builtin_amdgcn_s_wait_tensorcnt(0);
#else
  // Fallback: plain float4 streaming copy (still saturates HBM).
  const float4* s4 = (const float4*)src;
  float4* d4 = (float4*)dst;
  const int n4 = Cc * Hh * Ww / 4;
  for (int i = blockIdx.x * blockDim.x + threadIdx.x; i < n4;
       i += gridDim.x * blockDim.x)
    d4[i] = s4[i];
#endif
}

// ---------------------------------------------------------------------------
// Kernel 1: derive all 1000 window descriptors with the JAX PRNG.
// keys = split(key(seed), 1000); per window: kh,kw,kph,kpw,kperm = split(k,5)
// ---------------------------------------------------------------------------
__global__ void win_desc_kernel(const int* __restrict__ seedp,
                                u32* __restrict__ desc) {
  int w = blockIdx.x * blockDim.x + threadIdx.x;
  if (w >= NW) return;
  int64_t s = (int64_t)seedp[0];
  u32 r0 = (u32)(((uint64_t)s) >> 32);
  u32 r1 = (u32)((uint64_t)s);

  u32 k0, k1;
  threefry2x32(r0, r1, (u32)w, (u32)(w + NW), k0, k1);   // split(key, 1000)

  u32 kh0, kh1, kw0, kw1, kph0, kph1, kpw0, kpw1, kp0, kp1;
  threefry2x32(k0, k1, 0u, 5u, kh0, kh1);                // split(k, 5)
  threefry2x32(k0, k1, 1u, 6u, kw0, kw1);
  threefry2x32(k0, k1, 2u, 7u, kph0, kph1);
  threefry2x32(k0, k1, 3u, 8u, kpw0, kpw1);
  threefry2x32(k0, k1, 4u, 9u, kp0, kp1);

  u32 a, b;
  // randint(k,(),1,9): span=8 -> (2^16 % 8)==0 -> offset = bits & 7
  threefry2x32(kh0, kh1, 0u, 0u, a, b);
  int wsh = 1 + (int)(a & 7u);
  threefry2x32(kw0, kw1, 0u, 0u, a, b);
  int wsw = 1 + (int)(a & 7u);
  threefry2x32(kph0, kph1, 0u, 0u, a, b);
  int ph = (int)(u01(a) * (float)(Hh - wsh + 1));        // trunc == astype(i32)
  threefry2x32(kpw0, kpw1, 0u, 0u, a, b);
  int pw = (int)(u01(a) * (float)(Ww - wsw + 1));

  u32* d = desc + (size_t)w * 8;
  d[0] = (u32)ph | ((u32)pw << 16);
  d[1] = (u32)wsh | ((u32)wsw << 16);
  d[2] = kp0;
  d[3] = kp1;
  d[4] = 0; d[5] = 0; d[6] = 0; d[7] = 0;
}

// ---------------------------------------------------------------------------
// Kernel 2: one block per window, 192 threads = (c, m) over 3 x 64 patch.
// Destination of source (c,m) is the rank(uniform)-th window position;
// suppress the write iff any LATER window covers the destination pixel
// (last-writer-wins replaces the sequential scan).
// ---------------------------------------------------------------------------
__global__ void apply_windows(const float* __restrict__ orig,
                              float* __restrict__ out,
                              const u32* __restrict__ desc) {
  __shared__ u32 covA[NW];
  __shared__ u32 covB[NW];
  __shared__ float keyv[Cc * 64];

  int w = blockIdx.x;
  for (int i = threadIdx.x; i < NW; i += blockDim.x) {
    covA[i] = desc[(size_t)i * 8 + 0];
    covB[i] = desc[(size_t)i * 8 + 1];
  }
  __syncthreads();

  int tid = threadIdx.x;        // 0..191
  int c = tid >> 6;
  int m = tid & 63;

  u32 d0 = covA[w], d1 = covB[w];
  u32 kp0 = desc[(size_t)w * 8 + 2];
  u32 kp1 = desc[(size_t)w * 8 + 3];

  int ph = (int)(d0 & 0xFFFFu), pw = (int)(d0 >> 16);
  int wsh = (int)(d1 & 0xFFFFu), wsw = (int)(d1 >> 16);
  int psh = ph < (Hh - MB) ? ph : (Hh - MB);     // clip(ph, 0, H-8)
  int psw = pw < (Ww - MB) ? pw : (Ww - MB);
  int offh = ph - psh, offw = pw - psw;
  int mr = m >> 3, mc = m & 7;
  bool masked = (mr >= offh) && (mr < offh + wsh) &&
                (mc >= offw) && (mc < offw + wsw);

  // uniform(kperm, (3,64)): n=192, element i pairs with i+96 (iota halves)
  int i = c * 64 + m;
  u32 x0 = (i < 96) ? (u32)i : (u32)(i - 96);
  u32 o0, o1;
  threefry2x32(kp0, kp1, x0, x0 + 96u, o0, o1);
  u32 bits = (i < 96) ? o0 : o1;
  float key = masked ? u01(bits) : (float)(m + 2);   // "big" = m + 2.0
  keyv[tid] = key;
  __syncthreads();

  // stable ascending rank within channel (== jnp.argsort tie-break by index)
  int rank = 0;
  const float* kc = &keyv[c * 64];
#pragma unroll 8
  for (int j = 0; j < 64; ++j) {
    float kj = kc[j];
    rank += (kj < key || (kj == key && j < m)) ? 1 : 0;
  }

  if (masked) {
    // masked keys (<1) always rank before "big" keys (>=2): rank in [0, n)
    int dr = ph + rank / wsw;
    int dc = pw + rank % wsw;
    bool skip = false;
    for (int w2 = w + 1; w2 < NW; ++w2) {
      u32 a = covA[w2], b = covB[w2];
      int r0 = (int)(a & 0xFFFFu), c0 = (int)(a >> 16);
      int hh = (int)(b & 0xFFFFu), ww = (int)(b >> 16);
      if (dr >= r0 && dr < r0 + hh && dc >= c0 && dc < c0 + ww) {
        skip = true;   // a later window rewrites this pixel
        break;
      }
    }
    if (!skip) {
      size_t plane = (size_t)c * (size_t)Hh * (size_t)Ww;
      out[plane + (size_t)dr * Ww + (size_t)dc] =
          orig[plane + (size_t)(psh + mr) * Ww + (size_t)(psw + mc)];
    }
  }
}

// ---------------------------------------------------------------------------
extern "C" void kernel_launch(void* const* d_in, const int* in_sizes, int n_in,
                              void* d_out, int out_size, void* d_ws,
                              size_t ws_size, hipStream_t stream) {
  const float* image = (const float*)d_in[0];
  const int* seed = (const int*)d_in[1];
  float* out = (float*)d_out;
  u32* desc = (u32*)d_ws;                 // 1000 * 32 B = 32 KB scratch

  int nblk = (Cc * Hh) / TROWS;           // 3072 tiles, one wave each
  bulk_copy_tdm<<<nblk, 32, 0, stream>>>(image, out);
  win_desc_kernel<<<(NW + 255) / 256, 256, 0, stream>>>(seed, desc);
  apply_windows<<<NW, Cc * 64, 0, stream>>>(image, out, desc);
}